// ApplyWeights_9251359556349
// MI455X (gfx1250) — compile-verified
//
#include <hip/hip_runtime.h>
#include <stdint.h>

// out[b,p] = sum_k x[b, pix[k,p]] * weight[k,p]
//   B=8 batches, K=4 stencil, P_OUT=721*1440, N_IN=12*512*512
// Memory-bound gather; pix/weight staged to LDS via gfx1250 async copies.

#define TILE  256
#define KST   4
#define BATCH 8

// Generic pointers to LDS on amdgcn are {SHARED_BASE_hi32, lds_offset_lo32};
// the low 32 bits are exactly the wave-relative LDS byte address the async
// instructions consume in their VDST VGPR.
__device__ __forceinline__ uint32_t lds_off(const void* p) {
  return (uint32_t)(uintptr_t)p;
}

// gfx1250 async global->LDS copies (ASYNCcnt-tracked), GV addressing mode:
//   each lane copies from the global address in vaddr to the LDS address in vdst
__device__ __forceinline__ void async_ld_b64(uint32_t lds, const void* g) {
  asm volatile("global_load_async_to_lds_b64 %0, %1, off"
               :: "v"(lds), "v"((uint64_t)(uintptr_t)g) : "memory");
}
__device__ __forceinline__ void async_ld_b32(uint32_t lds, const void* g) {
  asm volatile("global_load_async_to_lds_b32 %0, %1, off"
               :: "v"(lds), "v"((uint64_t)(uintptr_t)g) : "memory");
}
template <int N>
__device__ __forceinline__ void wait_async() {
  asm volatile("s_wait_asynccnt %0" :: "i"(N) : "memory");
}

__global__ __launch_bounds__(TILE)
void healpix_regrid_9251359556349(const float* __restrict__ x,
                                  const float* __restrict__ wgt,
                                  const long long* __restrict__ pix,
                                  float* __restrict__ out,
                                  int n_src, int p_out, int num_tiles)
{
  // Double-buffered staging: 2 * (4*256*8 + 4*256*4) = 24 KB LDS / block.
  __shared__ long long s_idx[2][KST][TILE];
  __shared__ float     s_w  [2][KST][TILE];

  const int lane    = threadIdx.x;
  const int t0      = blockIdx.x;
  const int tstride = gridDim.x;
  if (t0 >= num_tiles) return;

  // Stage one tile's pix (b64) + weight (b32) into LDS buffer `buf`.
  // Tail lanes clamp the source address (keeps EXEC uniform; value unused).
  auto stage = [&](int buf, int tile) {
    int p = tile * TILE + lane;
    if (p >= p_out) p = p_out - 1;
#pragma unroll
    for (int k = 0; k < KST; ++k) {
      async_ld_b64(lds_off(&s_idx[buf][k][lane]), &pix[(size_t)k * p_out + p]);
      async_ld_b32(lds_off(&s_w  [buf][k][lane]), &wgt[(size_t)k * p_out + p]);
    }
  };

  stage(0, t0);                 // prologue: 8 async ops outstanding
  int buf = 0;

  for (int tile = t0; tile < num_tiles; tile += tstride) {
    const int nxt = tile + tstride;
    if (nxt < num_tiles) {
      stage(buf ^ 1, nxt);      // prefetch next tile (8 more async ops)
      wait_async<8>();          // in-order retire => current tile resident
    } else {
      wait_async<0>();          // last tile for this block
    }

    const int p = tile * TILE + lane;
    if (p < p_out) {
      float acc[BATCH];
#pragma unroll
      for (int b = 0; b < BATCH; ++b) acc[b] = 0.0f;

#pragma unroll
      for (int k = 0; k < KST; ++k) {
        const int   idx = (int)s_idx[buf][k][lane];  // each thread reads only
        const float wk  = s_w[buf][k][lane];         // slots it staged itself
        const float* xp = x + idx;
#pragma unroll
        for (int b = 0; b < BATCH; ++b)              // 32 independent gathers
          acc[b] = fmaf(xp[(size_t)b * n_src], wk, acc[b]);
      }

#pragma unroll
      for (int b = 0; b < BATCH; ++b)                // coalesced, write-once:
        __builtin_nontemporal_store(acc[b], &out[(size_t)b * p_out + p]);
    }
    buf ^= 1;
  }
}

extern "C" void kernel_launch(void* const* d_in, const int* in_sizes, int n_in,
                              void* d_out, int out_size, void* d_ws, size_t ws_size,
                              hipStream_t stream) {
  const float*     x   = (const float*)d_in[0];      // [B, N_IN] f32
  const float*     wgt = (const float*)d_in[1];      // [K, P_OUT] f32
  const long long* pix = (const long long*)d_in[2];  // [K, P_OUT] i64
  float*           out = (float*)d_out;              // [B, P_OUT] f32

  const int N_SRC = in_sizes[0] / BATCH;   // 3,145,728
  const int P_OUT = in_sizes[1] / KST;     // 1,038,240
  const int num_tiles = (P_OUT + TILE - 1) / TILE;   // 4056
  const int blocks = (num_tiles + 3) / 4;            // 1014 -> 4 tiles/block

  hipLaunchKernelGGL(healpix_regrid_9251359556349,
                     dim3(blocks), dim3(TILE), 0, stream,
                     x, wgt, pix, out, N_SRC, P_OUT, num_tiles);
}